// GraphSAGE_26603027431847
// MI455X (gfx1250) — compile-verified
//
#include <hip/hip_runtime.h>
#include <math.h>

typedef __attribute__((ext_vector_type(2))) float v2f;
typedef __attribute__((ext_vector_type(8))) float v8f;

// ---------------------------------------------------------------------------
// Zero fill
// ---------------------------------------------------------------------------
__global__ void zero_f32(float* __restrict__ p, int n) {
  int i = blockIdx.x * blockDim.x + threadIdx.x;
  if (i < n) p[i] = 0.0f;
}

// ---------------------------------------------------------------------------
// Edge scatter: one wave32 per edge, 2 channels per lane (64-wide features).
// Atomic fp32 adds resolve in L2 (accumulator is 25.6 MB << 192 MB L2).
// ---------------------------------------------------------------------------
__global__ void scatter_edges(const float* __restrict__ feat,
                              const int*  __restrict__ src,
                              const int*  __restrict__ dst,
                              float* __restrict__ agg,
                              float* __restrict__ cnt,
                              int E, int addCount) {
  int gid  = blockIdx.x * blockDim.x + threadIdx.x;
  int e    = gid >> 5;
  int lane = gid & 31;
  if (e >= E) return;
  int s = src[e];
  int d = dst[e];
  long so = (long)s * 64;
  long dofs = (long)d * 64;
  float v0 = feat[so + lane];
  float v1 = feat[so + 32 + lane];
  unsafeAtomicAdd(&agg[dofs + lane], v0);
  unsafeAtomicAdd(&agg[dofs + 32 + lane], v1);
  if (addCount && lane == 0) unsafeAtomicAdd(&cnt[d], 1.0f);
}

// cnt[i] -> 1 / max(cnt[i], 1)   (PyG 'mean' clamped-degree divide)
__global__ void recip_cnt(float* __restrict__ cnt, int n) {
  int i = blockIdx.x * blockDim.x + threadIdx.x;
  if (i < n) cnt[i] = 1.0f / fmaxf(cnt[i], 1.0f);
}

// ---------------------------------------------------------------------------
// Fused SAGE linear stage, one wave per 16-node x 16-channel output tile:
//   out = agg_sum*invdeg @ Wl^T + bias + xin @ Wr^T   [+ BN + ReLU]
// K = 64 walked with V_WMMA_F32_16X16X4_F32 (fp32 in / fp32 accum).
// A 16x4 layout: lane L -> row L&15, K-pair 2*(L>>4).  B 4x16 symmetric.
// D 16x16 layout: VGPR j -> M = j + 8*(lane>=16), N = lane&15.
// Row guards via selects so EXEC stays all-1s for WMMA.
// ---------------------------------------------------------------------------
__global__ void sage_gemm(const float* __restrict__ aggsum,
                          const float* __restrict__ cntR,
                          const float* __restrict__ xin,
                          const float* __restrict__ Wl,   // [outC x 64] row-major
                          const float* __restrict__ Wr,   // [outC x 64] row-major
                          const float* __restrict__ bias, // [outC]
                          const float* __restrict__ bn_g,
                          const float* __restrict__ bn_b,
                          const float* __restrict__ bn_m,
                          const float* __restrict__ bn_v,
                          float* __restrict__ out,
                          int N, int outC, int doBN) {
  const int lane     = threadIdx.x & 31;
  const int wave     = threadIdx.x >> 5;
  const int row      = lane & 15;   // A row within tile; also B column index
  const int khalf    = lane >> 4;   // which K pair / which D half
  const int nodeBase = blockIdx.x * 16;
  const int colBase  = wave * 16;

  const int  node  = nodeBase + row;
  const bool rowOK = node < N;
  const long aoff  = (long)(rowOK ? node : 0) * 64;
  const float invc = rowOK ? cntR[node] : 0.0f;
  const float rowm = rowOK ? 1.0f : 0.0f;
  const long  woff = (long)(colBase + row) * 64;  // weight row for this lane's B col

  v8f acc = {};
  #pragma unroll
  for (int k0 = 0; k0 < 64; k0 += 4) {
    const int kb = k0 + 2 * khalf;
    v2f a, b;
    // lin_l on the mean-aggregated neighborhood
    a.x = aggsum[aoff + kb]     * invc;
    a.y = aggsum[aoff + kb + 1] * invc;
    b.x = Wl[woff + kb];
    b.y = Wl[woff + kb + 1];
    acc = __builtin_amdgcn_wmma_f32_16x16x4_f32(false, a, false, b,
                                                (short)0, acc, false, false);
    // lin_r on the node's own features
    a.x = xin[aoff + kb]     * rowm;
    a.y = xin[aoff + kb + 1] * rowm;
    b.x = Wr[woff + kb];
    b.y = Wr[woff + kb + 1];
    acc = __builtin_amdgcn_wmma_f32_16x16x4_f32(false, a, false, b,
                                                (short)0, acc, false, false);
  }

  // Epilogue: bias (+ BatchNorm eval + ReLU), channel fixed per lane.
  const int c  = colBase + row;
  const float bs = bias[c];
  float g = 1.0f, bb = 0.0f, mn = 0.0f, iv = 1.0f;
  if (doBN) {
    g  = bn_g[c];
    bb = bn_b[c];
    mn = bn_m[c];
    iv = rsqrtf(bn_v[c] + 1e-5f);
  }
  #pragma unroll
  for (int j = 0; j < 8; ++j) {
    const int m = j + 8 * khalf;
    const int onode = nodeBase + m;
    if (onode >= N) continue;
    float v = acc[j] + bs;
    if (doBN) {
      v = (v - mn) * (g * iv) + bb;
      v = fmaxf(v, 0.0f);
    }
    out[(long)onode * outC + c] = v;
  }
}

// ---------------------------------------------------------------------------
// log_softmax over 32 classes: one wave per node, one class per lane,
// butterfly shuffle reductions for max and sum-exp.
// ---------------------------------------------------------------------------
__global__ void log_softmax32(float* __restrict__ out, int N) {
  int gid  = blockIdx.x * blockDim.x + threadIdx.x;
  int node = gid >> 5;
  int lane = gid & 31;
  if (node >= N) return;
  long o = (long)node * 32 + lane;
  float v = out[o];
  float m = v;
  #pragma unroll
  for (int off = 16; off > 0; off >>= 1)
    m = fmaxf(m, __shfl_xor(m, off, 32));
  float e = __expf(v - m);
  float s = e;
  #pragma unroll
  for (int off = 16; off > 0; off >>= 1)
    s += __shfl_xor(s, off, 32);
  out[o] = v - m - __logf(s);
}

// ---------------------------------------------------------------------------
extern "C" void kernel_launch(void* const* d_in, const int* in_sizes, int n_in,
                              void* d_out, int out_size, void* d_ws, size_t ws_size,
                              hipStream_t stream) {
  const float* x    = (const float*)d_in[0];
  const int*   ei   = (const int*)d_in[1];
  const float* W_l0 = (const float*)d_in[2];
  const float* b_l0 = (const float*)d_in[3];
  const float* W_r0 = (const float*)d_in[4];
  const float* bn_g = (const float*)d_in[5];
  const float* bn_b = (const float*)d_in[6];
  const float* bn_m = (const float*)d_in[7];
  const float* bn_v = (const float*)d_in[8];
  const float* W_l1 = (const float*)d_in[9];
  const float* b_l1 = (const float*)d_in[10];
  const float* W_r1 = (const float*)d_in[11];
  float* outp = (float*)d_out;

  const int N = in_sizes[0] / 64;   // 100000
  const int E = in_sizes[1] / 2;    // 1600000
  const int HID = 64, OUT = 32;

  // Workspace: [cnt N][agg N*64][h N*64]  (~51.6 MB); agg reused by layer 1.
  float* cnt = (float*)d_ws;
  float* agg = cnt + N;
  float* h   = agg + (size_t)N * 64;

  const int* src = ei;
  const int* dst = ei + E;

  // Layer 0 aggregation
  {
    int n = N * 65;  // cnt + agg are contiguous
    zero_f32<<<(n + 255) / 256, 256, 0, stream>>>(cnt, n);
  }
  {
    long t = (long)E * 32;
    scatter_edges<<<(unsigned)((t + 255) / 256), 256, 0, stream>>>(
        x, src, dst, agg, cnt, E, 1);
  }
  recip_cnt<<<(N + 255) / 256, 256, 0, stream>>>(cnt, N);

  // Layer 0 fused GEMM + BN + ReLU -> h   (4 col tiles => 128 threads)
  sage_gemm<<<(N + 15) / 16, 128, 0, stream>>>(
      agg, cnt, x, W_l0, W_r0, b_l0, bn_g, bn_b, bn_m, bn_v, h, N, HID, 1);

  // Layer 1 aggregation (degree reciprocals unchanged)
  {
    int n = N * 64;
    zero_f32<<<(n + 255) / 256, 256, 0, stream>>>(agg, n);
  }
  {
    long t = (long)E * 32;
    scatter_edges<<<(unsigned)((t + 255) / 256), 256, 0, stream>>>(
        h, src, dst, agg, cnt, E, 0);
  }

  // Layer 1 GEMM -> logits in d_out   (2 col tiles => 64 threads)
  sage_gemm<<<(N + 15) / 16, 64, 0, stream>>>(
      agg, cnt, h, W_l1, W_r1, b_l1, nullptr, nullptr, nullptr, nullptr,
      outp, N, OUT, 0);

  // In-place log_softmax over the 32 classes
  {
    long t = (long)N * 32;
    log_softmax32<<<(unsigned)((t + 255) / 256), 256, 0, stream>>>(outp, N);
  }
}